// SS2D_19542101197650
// MI455X (gfx1250) — compile-verified
//
#include <hip/hip_runtime.h>
#include <hip/hip_bf16.h>
#include <math.h>

// ---------------- problem constants (match reference) ----------------
#define BATCH   4
#define HD      64
#define WD      64
#define DMODEL  128
#define DINNER  256
#define NSTATE  16
#define KDIR    4
#define RRANK   8
#define LLEN    (HD * WD)            // 4096
#define CPROJ   (RRANK + 2 * NSTATE) // 40
#define LN_EPS  1e-5f

// chunked scan parameters
#define NCH     16                   // chunks along L
#define CH      (LLEN / NCH)         // 256 steps per chunk

// CDNA5 async global->LDS path (guarded; falls back to VGPR round-trip)
#if defined(__AMDGCN__) && \
    __has_builtin(__builtin_amdgcn_global_load_async_to_lds_b128) && \
    __has_builtin(__builtin_amdgcn_s_wait_asynccnt)
#define USE_ASYNC_LDS 1
#else
#define USE_ASYNC_LDS 0
#endif

typedef __attribute__((ext_vector_type(2))) float v2f;
typedef __attribute__((ext_vector_type(8))) float v8f;

#if USE_ASYNC_LDS
typedef int v4i __attribute__((vector_size(16)));
typedef __attribute__((address_space(1))) v4i gv4i;   // global 16B vector
typedef __attribute__((address_space(3))) v4i lv4i;   // LDS 16B vector
#endif

// scan-order index map: scan position l (direction k) -> spatial index h*WD+w
__device__ __forceinline__ int scan_map(int k, int l) {
  int ll = (k >= 2) ? (LLEN - 1 - l) : l;
  if (k & 1) {
    // w-major order: l = w*HD + h  ->  spatial = h*WD + w
    return ((ll & (HD - 1)) << 6) | (ll >> 6);
  }
  return ll;
}

__device__ __forceinline__ float silu_f(float v) {
  return v / (1.0f + __expf(-v));
}

__device__ __forceinline__ float softplus_f(float v) {
  return (v > 20.0f) ? v : log1pf(__expf(v));
}

// ---------------- K0: zero the merge accumulator ----------------
__global__ void k_zero(float* __restrict__ p, int n) {
  int t = blockIdx.x * blockDim.x + threadIdx.x;
  if (t < n) p[t] = 0.0f;
}

// ---------------- K1: in_proj GEMM (WMMA f32 16x16x4) ----------------
// xz[m,n] = sum_k x[m,k] * W[n,k];  m in [0,16384), n in [0,512)
__global__ void k_inproj(const float* __restrict__ x, const float* __restrict__ w,
                         float* __restrict__ xh, float* __restrict__ z) {
  const int lane = threadIdx.x & 31;
  const int half = lane >> 4;
  const int idx  = lane & 15;
  const int m0 = blockIdx.x * 16;
  const int n0 = blockIdx.y * 16;

  v8f acc = {};
  const float* arow = x + (size_t)(m0 + idx) * DMODEL;
  const float* brow = w + (size_t)(n0 + idx) * DMODEL;
  for (int k0 = 0; k0 < DMODEL; k0 += 4) {
    v2f a, b;
    a.x = arow[k0 + 2 * half];
    a.y = arow[k0 + 2 * half + 1];
    b.x = brow[k0 + 2 * half];
    b.y = brow[k0 + 2 * half + 1];
    acc = __builtin_amdgcn_wmma_f32_16x16x4_f32(false, a, false, b,
                                                (short)0, acc, false, false);
  }
#pragma unroll
  for (int j = 0; j < 8; ++j) {
    int m = m0 + j + 8 * half;
    int n = n0 + idx;
    float v = acc[j];
    int b_ = m / LLEN;
    int l  = m % LLEN;
    if (n < DINNER) {
      xh[((size_t)b_ * DINNER + n) * LLEN + l] = v;
    } else {
      z[(size_t)m * DINNER + (n - DINNER)] = silu_f(v);
    }
  }
}

// ---------------- K2: depthwise 3x3 conv + bias + SiLU ----------------
__global__ void k_dwconv(const float* __restrict__ xh, const float* __restrict__ cw,
                         const float* __restrict__ cb,
                         float* __restrict__ xc_d, float* __restrict__ xc_l) {
  int t = blockIdx.x * blockDim.x + threadIdx.x;
  if (t >= BATCH * DINNER * LLEN) return;
  int ww = t & (WD - 1);
  int hh = (t >> 6) & (HD - 1);
  int d  = (t / LLEN) % DINNER;
  int b  = t / (LLEN * DINNER);

  const float* wp = cw + d * 9;
  const float* xp = xh + ((size_t)b * DINNER + d) * LLEN;
  float acc = cb[d];
#pragma unroll
  for (int dy = 0; dy < 3; ++dy) {
#pragma unroll
    for (int dx = 0; dx < 3; ++dx) {
      int y = hh + dy - 1, x2 = ww + dx - 1;
      if (y >= 0 && y < HD && x2 >= 0 && x2 < WD)
        acc += wp[dy * 3 + dx] * xp[y * WD + x2];
    }
  }
  acc = silu_f(acc);
  int l = hh * WD + ww;
  xc_d[((size_t)b * DINNER + d) * LLEN + l] = acc;
  xc_l[((size_t)b * LLEN + l) * DINNER + d] = acc;
}

// ---------------- K3: x_proj GEMM (WMMA f32 16x16x4) ----------------
// x_dbl[b,k,c,l] = sum_d xpw[k,c,d] * xc[b,d,scan_map(k,l)]
__global__ void k_xproj(const float* __restrict__ xc_d, const float* __restrict__ xpw,
                        float* __restrict__ x_dbl) {
  const int lane = threadIdx.x & 31;
  const int half = lane >> 4;
  const int idx  = lane & 15;
  const int l0 = blockIdx.x * 16;
  const int c0 = blockIdx.y * 16;
  const int bk = blockIdx.z;
  const int b  = bk / KDIR;
  const int k  = bk % KDIR;

  const int c  = c0 + idx;
  const int sp = scan_map(k, l0 + idx);
  const float* abase = xpw + ((size_t)k * CPROJ + c) * DINNER;
  const float* bbase = xc_d + (size_t)b * DINNER * LLEN + sp;

  v8f acc = {};
  for (int d0 = 0; d0 < DINNER; d0 += 4) {
    v2f a, bb;
    int ka = d0 + 2 * half;
    a.x = (c < CPROJ) ? abase[ka]     : 0.0f;
    a.y = (c < CPROJ) ? abase[ka + 1] : 0.0f;
    bb.x = bbase[(size_t)ka * LLEN];
    bb.y = bbase[(size_t)(ka + 1) * LLEN];
    acc = __builtin_amdgcn_wmma_f32_16x16x4_f32(false, a, false, bb,
                                                (short)0, acc, false, false);
  }
#pragma unroll
  for (int j = 0; j < 8; ++j) {
    int cc = c0 + j + 8 * half;
    int l  = l0 + idx;
    if (cc < CPROJ)
      x_dbl[((size_t)bk * CPROJ + cc) * LLEN + l] = acc[j];
  }
}

// ---- LDS tile staging: async DMA engine if available, else VGPR round-trip.
// nrows*CH floats, rows contiguous (16B-aligned), from xdb (row stride LLEN).
__device__ __forceinline__ void stage_tile(float* sx, const float* xdb, int nrows) {
#if USE_ASYNC_LDS
  for (int i = threadIdx.x; i < (nrows * CH) / 4; i += blockDim.x) {
    const int e   = i * 4;
    const int row = e / CH, col = e % CH;
    __builtin_amdgcn_global_load_async_to_lds_b128(
        (gv4i*)(xdb + (size_t)row * LLEN + col),
        (lv4i*)(&sx[e]),
        0, 0);
  }
  __builtin_amdgcn_s_wait_asynccnt(0);
#else
  for (int i = threadIdx.x; i < nrows * CH; i += blockDim.x) {
    int row = i / CH, col = i % CH;
    sx[i] = xdb[(size_t)row * LLEN + col];
  }
#endif
  __syncthreads();
}

// ============ chunked parallel selective scan (3 passes) ============
// Recurrence per (b,k,d,n): h_l = exp(dt_l*A_n)*h_{l-1} + dt_l*u_l*B_n(l)
// Chunk transition: h_end = exp(A_n * sum(dt)) * h_start + q  (q = local scan from 0)

// ---- Pass A: per-chunk local states q and dt-sums, starting from h=0 ----
__global__ void k_scan_a(const float* __restrict__ xc_l, const float* __restrict__ x_dbl,
                         const float* __restrict__ dtw, const float* __restrict__ dtb,
                         const float* __restrict__ A_logs,
                         float* __restrict__ hout, float* __restrict__ sdt) {
  __shared__ float sx[(RRANK + NSTATE) * CH];   // 24*256 floats = 24 KB

  const int d  = threadIdx.x;
  const int c  = blockIdx.x % NCH;
  const int bk = blockIdx.x / NCH;
  const int b  = bk / KDIR;
  const int k  = bk % KDIR;
  const int kd = k * DINNER + d;
  const int l0 = c * CH;

  const float* xdb = x_dbl + (size_t)bk * CPROJ * LLEN + l0;
  stage_tile(sx, xdb, RRANK + NSTATE);          // rows 0..23 (dt rank + B)

  float A[NSTATE];
#pragma unroll
  for (int n = 0; n < NSTATE; ++n)
    A[n] = -__expf(A_logs[(size_t)kd * NSTATE + n]);
  float wr[RRANK];
#pragma unroll
  for (int r = 0; r < RRANK; ++r)
    wr[r] = dtw[(size_t)kd * RRANK + r];
  const float bias = dtb[kd];

  float h[NSTATE];
#pragma unroll
  for (int n = 0; n < NSTATE; ++n) h[n] = 0.0f;
  float dsum = 0.0f;

  const float* xcb = xc_l + (size_t)b * LLEN * DINNER + d;

  for (int j = 0; j < CH; ++j) {
    const int sp = scan_map(k, l0 + j);
    const float u = xcb[(size_t)sp * DINNER];
    float dt = bias;
#pragma unroll
    for (int r = 0; r < RRANK; ++r) dt += wr[r] * sx[r * CH + j];
    dt = softplus_f(dt);
    dsum += dt;
    const float du = dt * u;
#pragma unroll
    for (int n = 0; n < NSTATE; ++n)
      h[n] = __expf(dt * A[n]) * h[n] + du * sx[(RRANK + n) * CH + j];
  }

  const size_t base = ((size_t)blockIdx.x * DINNER + d) * NSTATE;
#pragma unroll
  for (int n = 0; n < NSTATE; ++n) hout[base + n] = h[n];
  sdt[(size_t)blockIdx.x * DINNER + d] = dsum;
}

// ---- Pass B: serial chunk-state composition (tiny) ----
__global__ void k_chain(const float* __restrict__ hout, const float* __restrict__ sdt,
                        const float* __restrict__ A_logs, float* __restrict__ hinit) {
  int t = blockIdx.x * blockDim.x + threadIdx.x;
  if (t >= BATCH * KDIR * DINNER) return;
  const int d  = t % DINNER;
  const int bk = t / DINNER;
  const int k  = bk % KDIR;
  const int kd = k * DINNER + d;

  float A[NSTATE];
#pragma unroll
  for (int n = 0; n < NSTATE; ++n)
    A[n] = -__expf(A_logs[(size_t)kd * NSTATE + n]);

  float h[NSTATE];
#pragma unroll
  for (int n = 0; n < NSTATE; ++n) h[n] = 0.0f;

  for (int c = 0; c < NCH; ++c) {
    const size_t base = (((size_t)bk * NCH + c) * DINNER + d) * NSTATE;
#pragma unroll
    for (int n = 0; n < NSTATE; ++n) hinit[base + n] = h[n];
    const float S = sdt[((size_t)bk * NCH + c) * DINNER + d];
#pragma unroll
    for (int n = 0; n < NSTATE; ++n)
      h[n] = __expf(A[n] * S) * h[n] + hout[base + n];
  }
}

// ---- Pass C: re-run each chunk from its true initial state, emit y ----
__global__ void k_scan_c(const float* __restrict__ xc_l, const float* __restrict__ x_dbl,
                         const float* __restrict__ dtw, const float* __restrict__ dtb,
                         const float* __restrict__ A_logs, const float* __restrict__ Ds,
                         const float* __restrict__ hinit, float* __restrict__ ym) {
  __shared__ float sx[CPROJ * CH];              // 40*256 floats = 40 KB

  const int d  = threadIdx.x;
  const int c  = blockIdx.x % NCH;
  const int bk = blockIdx.x / NCH;
  const int b  = bk / KDIR;
  const int k  = bk % KDIR;
  const int kd = k * DINNER + d;
  const int l0 = c * CH;

  const float* xdb = x_dbl + (size_t)bk * CPROJ * LLEN + l0;
  stage_tile(sx, xdb, CPROJ);                   // all 40 rows

  float A[NSTATE];
#pragma unroll
  for (int n = 0; n < NSTATE; ++n)
    A[n] = -__expf(A_logs[(size_t)kd * NSTATE + n]);
  float wr[RRANK];
#pragma unroll
  for (int r = 0; r < RRANK; ++r)
    wr[r] = dtw[(size_t)kd * RRANK + r];
  const float bias = dtb[kd];
  const float Dsk  = Ds[kd];

  float h[NSTATE];
  const size_t hbase = ((size_t)blockIdx.x * DINNER + d) * NSTATE;
#pragma unroll
  for (int n = 0; n < NSTATE; ++n) h[n] = hinit[hbase + n];

  const float* xcb = xc_l + (size_t)b * LLEN * DINNER + d;
  float* ymb = ym + (size_t)b * LLEN * DINNER + d;

  for (int j = 0; j < CH; ++j) {
    const int sp = scan_map(k, l0 + j);
    // prefetch the u stream ~8 steps ahead (strided for w-major directions)
    if (j + 8 < CH) {
      const int spn = scan_map(k, l0 + j + 8);
      __builtin_prefetch(xcb + (size_t)spn * DINNER, 0, 0);
    }
    const float u = xcb[(size_t)sp * DINNER];
    float dt = bias;
#pragma unroll
    for (int r = 0; r < RRANK; ++r) dt += wr[r] * sx[r * CH + j];
    dt = softplus_f(dt);
    const float du = dt * u;
    float y = 0.0f;
#pragma unroll
    for (int n = 0; n < NSTATE; ++n) {
      h[n] = __expf(dt * A[n]) * h[n] + du * sx[(RRANK + n) * CH + j];
      y += h[n] * sx[(RRANK + NSTATE + n) * CH + j];
    }
    y += Dsk * u;
    atomicAdd(ymb + (size_t)sp * DINNER, y);
  }
}

// ---------------- K5: layernorm over DI + multiply by SiLU(z) ----------------
__global__ void k_norm(const float* __restrict__ ym, const float* __restrict__ z,
                       const float* __restrict__ gw, const float* __restrict__ gb,
                       float* __restrict__ yn) {
  const int row = blockIdx.x;      // (b*L + l)
  const int d   = threadIdx.x;     // channel
  __shared__ float red[DINNER];

  float v = ym[(size_t)row * DINNER + d];
  red[d] = v;
  __syncthreads();
  for (int s = DINNER / 2; s > 0; s >>= 1) {
    if (d < s) red[d] += red[d + s];
    __syncthreads();
  }
  const float mu = red[0] * (1.0f / DINNER);
  __syncthreads();
  float cdiff = v - mu;
  red[d] = cdiff * cdiff;
  __syncthreads();
  for (int s = DINNER / 2; s > 0; s >>= 1) {
    if (d < s) red[d] += red[d + s];
    __syncthreads();
  }
  const float var = red[0] * (1.0f / DINNER);
  const float inv = rsqrtf(var + LN_EPS);
  float o = cdiff * inv * gw[d] + gb[d];
  o *= z[(size_t)row * DINNER + d];   // z already SiLU'd in K1
  yn[(size_t)row * DINNER + d] = o;
}

// ---------------- K6: out_proj GEMM (WMMA f32 16x16x4) ----------------
__global__ void k_outproj(const float* __restrict__ yn, const float* __restrict__ opw,
                          float* __restrict__ out) {
  const int lane = threadIdx.x & 31;
  const int half = lane >> 4;
  const int idx  = lane & 15;
  const int m0 = blockIdx.x * 16;
  const int n0 = blockIdx.y * 16;

  const float* arow = yn  + (size_t)(m0 + idx) * DINNER;
  const float* brow = opw + (size_t)(n0 + idx) * DINNER;
  v8f acc = {};
  for (int k0 = 0; k0 < DINNER; k0 += 4) {
    v2f a, b;
    a.x = arow[k0 + 2 * half];
    a.y = arow[k0 + 2 * half + 1];
    b.x = brow[k0 + 2 * half];
    b.y = brow[k0 + 2 * half + 1];
    acc = __builtin_amdgcn_wmma_f32_16x16x4_f32(false, a, false, b,
                                                (short)0, acc, false, false);
  }
#pragma unroll
  for (int j = 0; j < 8; ++j) {
    int m = m0 + j + 8 * half;
    int n = n0 + idx;
    out[(size_t)m * DMODEL + n] = acc[j];
  }
}

// ---------------- host launch ----------------
extern "C" void kernel_launch(void* const* d_in, const int* in_sizes, int n_in,
                              void* d_out, int out_size, void* d_ws, size_t ws_size,
                              hipStream_t stream) {
  (void)in_sizes; (void)n_in; (void)out_size; (void)ws_size;

  const float* x      = (const float*)d_in[0];   // (B,H,W,DM)
  const float* ipw    = (const float*)d_in[1];   // (2*DI, DM)
  const float* convw  = (const float*)d_in[2];   // (DI,1,3,3)
  const float* convb  = (const float*)d_in[3];   // (DI,)
  const float* xpw    = (const float*)d_in[4];   // (K,40,DI)
  const float* dtw    = (const float*)d_in[5];   // (K,DI,R)
  const float* dtb    = (const float*)d_in[6];   // (K,DI)
  const float* A_logs = (const float*)d_in[7];   // (K*DI,N)
  const float* Dsv    = (const float*)d_in[8];   // (K*DI,)
  const float* gnw    = (const float*)d_in[9];   // (DI,)
  const float* gnb    = (const float*)d_in[10];  // (DI,)
  const float* opw    = (const float*)d_in[11];  // (DM,DI)
  float* out = (float*)d_out;

  // workspace layout (floats)
  const size_t S1   = (size_t)BATCH * DINNER * LLEN;               // 4,194,304
  const size_t SDBL = (size_t)BATCH * KDIR * CPROJ * LLEN;         // 2,621,440
  const size_t SST  = (size_t)BATCH * KDIR * NCH * DINNER * NSTATE;// 1,048,576

  float* ws    = (float*)d_ws;
  float* xh    = ws;                       // (B,DI,L)   pre-conv
  float* zbuf  = ws + S1;                  // (B*L,DI)   SiLU(z)
  float* xc_d  = ws + 2 * S1;              // (B,DI,L)   post-conv, d-major
  float* xc_l  = ws + 3 * S1;              // (B,L,DI)   post-conv, l-major
  float* x_dbl = ws + 4 * S1;              // (B*K,40,L)
  float* ym    = ws + 4 * S1 + SDBL;       // (B*L,DI)   merge accumulator
  float* hout  = ym + S1;                  // (B*K*NCH, DI, N) chunk-local states
  float* hinit = hout + SST;               // (B*K*NCH, DI, N) chunk init states
  float* sdtb  = hinit + SST;              // (B*K*NCH, DI)    chunk dt sums
  float* yn    = xh;                       // reuse xh after conv

  // K0: zero merge accumulator
  k_zero<<<(int)((S1 + 255) / 256), 256, 0, stream>>>(ym, (int)S1);

  // K1: in_proj (M=16384, N=512, K=128)
  k_inproj<<<dim3(1024, 32, 1), 32, 0, stream>>>(x, ipw, xh, zbuf);

  // K2: depthwise conv + SiLU
  {
    int n = BATCH * DINNER * LLEN;
    k_dwconv<<<(n + 255) / 256, 256, 0, stream>>>(xh, convw, convb, xc_d, xc_l);
  }

  // K3: x_proj per direction (C=40 padded to 48)
  k_xproj<<<dim3(LLEN / 16, 3, BATCH * KDIR), 32, 0, stream>>>(xc_d, xpw, x_dbl);

  // K4a/b/c: chunked parallel selective scan
  k_scan_a<<<BATCH * KDIR * NCH, DINNER, 0, stream>>>(xc_l, x_dbl, dtw, dtb,
                                                      A_logs, hout, sdtb);
  k_chain<<<(BATCH * KDIR * DINNER + 255) / 256, 256, 0, stream>>>(hout, sdtb,
                                                                   A_logs, hinit);
  k_scan_c<<<BATCH * KDIR * NCH, DINNER, 0, stream>>>(xc_l, x_dbl, dtw, dtb,
                                                      A_logs, Dsv, hinit, ym);

  // K5: layernorm + gate
  k_norm<<<BATCH * LLEN, DINNER, 0, stream>>>(ym, zbuf, gnw, gnb, yn);

  // K6: out_proj (M=16384, N=128, K=256)
  k_outproj<<<dim3(1024, 8, 1), 32, 0, stream>>>(yn, opw, out);
}